// Loss_yolov1_30717606101349
// MI455X (gfx1250) — compile-verified
//
#include <hip/hip_runtime.h>

#define L_COORD 5.0f
#define L_NOOBJ 0.5f

typedef float v2f_ __attribute__((ext_vector_type(2)));
typedef float v8f_ __attribute__((ext_vector_type(8)));

// Exact 32-lane f32 sum using the matrix pipe:
// A = ones(16x4), B holds one partial per lane (B VGPR1 = 0), C = 0.
// D[m][n] = sum_k B[k][n]  -> 16 column sums (replicated in both 16-lane
// halves of D VGPR0); each lane's value is counted exactly once across the
// columns, so 4 xor-shuffle rounds over the 16-lane half yield the full sum.
static __device__ __forceinline__ float wave_sum32(float v) {
    v2f_ a; a[0] = 1.0f; a[1] = 1.0f;   // A = ones
    v2f_ b; b[0] = v;    b[1] = 0.0f;   // per-lane partial occupies one B slot
    v8f_ c = {};
    c = __builtin_amdgcn_wmma_f32_16x16x4_f32(
            /*neg_a=*/false, a, /*neg_b=*/false, b,
            /*c_mod=*/(short)0, c, /*reuse_a=*/false, /*reuse_b=*/false);
    float x = c[0];
    x += __shfl_xor(x, 1);
    x += __shfl_xor(x, 2);
    x += __shfl_xor(x, 4);
    x += __shfl_xor(x, 8);
    return x;   // every lane now holds the full 32-lane sum
}

static __device__ __forceinline__ float iou5(float b0, float b1, float b2, float b3,
                                             float t0, float t1, float t2, float t3) {
    float ax1 = b0 - 0.5f * b2, ay1 = b1 - 0.5f * b3;
    float ax2 = b0 + 0.5f * b2, ay2 = b1 + 0.5f * b3;
    float bx1 = t0 - 0.5f * t2, by1 = t1 - 0.5f * t3;
    float bx2 = t0 + 0.5f * t2, by2 = t1 + 0.5f * t3;
    float iw = fmaxf(fminf(ax2, bx2) - fmaxf(ax1, bx1), 0.0f);
    float ih = fmaxf(fminf(ay2, by2) - fmaxf(ay1, by1), 0.0f);
    float inter = iw * ih;
    float uni = b2 * b3 + t2 * t3 - inter;
    return inter / fmaxf(uni, 1e-10f);
}

__global__ __launch_bounds__(256)
void yolo_partial(const float* __restrict__ pred,
                  const float* __restrict__ targ,
                  float* __restrict__ part, int cells) {
    float loc = 0.f, cobj = 0.f, cnoobj = 0.f, clss = 0.f, nobj = 0.f;
    const int stride = gridDim.x * blockDim.x;

    for (int idx = blockIdx.x * blockDim.x + threadIdx.x; idx < cells; idx += stride) {
        // prefetch next stride's cell (global_prefetch_b8)
        int nidx = idx + stride;
        if (nidx < cells) {
            __builtin_prefetch(pred + (size_t)nidx * 30, 0, 1);
            __builtin_prefetch(targ + (size_t)nidx * 30, 0, 1);
        }

        const v2f_* P = (const v2f_*)(pred + (size_t)idx * 30);
        const v2f_* T = (const v2f_*)(targ + (size_t)idx * 30);
        float p[30], t[30];
#pragma unroll
        for (int i = 0; i < 15; ++i) { v2f_ x = P[i]; p[2*i] = x[0]; p[2*i+1] = x[1]; }
#pragma unroll
        for (int i = 0; i < 15; ++i) { v2f_ x = T[i]; t[2*i] = x[0]; t[2*i+1] = x[1]; }

        float objf = (t[4] == 1.0f) ? 1.0f : 0.0f;

        float iou0 = iou5(p[0], p[1], p[2], p[3], t[0], t[1], t[2], t[3]);
        float iou1 = iou5(p[5], p[6], p[7], p[8], t[0], t[1], t[2], t[3]);
        bool pick0 = iou0 > iou1;                 // ties pick box 1 (matches ref)
        float w0 = pick0 ? p[0] : p[5];
        float w1 = pick0 ? p[1] : p[6];
        float w2 = pick0 ? p[2] : p[7];
        float w3 = pick0 ? p[3] : p[8];
        float w4 = pick0 ? p[4] : p[9];
        float wiou = fmaxf(iou0, iou1);

        float dx = w0 - t[0], dy = w1 - t[1];
        float sw = sqrtf(w2) - sqrtf(t[2]);
        float sh = sqrtf(w3) - sqrtf(t[3]);
        loc += (dx * dx + dy * dy + sw * sw + sh * sh) * objf;

        float dc = w4 - wiou;
        cobj += dc * dc * objf;

        // log-softmax over p[10:30] + CE at target argmax (first-max semantics)
        float m = p[10];
#pragma unroll
        for (int j = 11; j < 30; ++j) m = fmaxf(m, p[j]);
        float s = 0.0f;
#pragma unroll
        for (int j = 10; j < 30; ++j) s += expf(p[j] - m);
        float bv = t[10], pv = p[10];
#pragma unroll
        for (int j = 11; j < 30; ++j) {
            bool upd = t[j] > bv;                 // strict > keeps first max
            bv = upd ? t[j] : bv;
            pv = upd ? p[j] : pv;
        }
        float ce = -(pv - m - logf(s));
        clss += ce * objf;
        nobj += objf;

        float d4 = p[4] - t[4], d9 = p[9] - t[9];
        cnoobj += (d4 * d4 + d9 * d9) * (1.0f - objf);
    }

    // EXEC is all-ones here (loop reconverged, no divergence yet): WMMA-legal.
    float r[5];
    r[0] = wave_sum32(loc);
    r[1] = wave_sum32(cobj);
    r[2] = wave_sum32(cnoobj);
    r[3] = wave_sum32(clss);
    r[4] = wave_sum32(nobj);

    __shared__ float sm[8 * 5];
    const int lane = threadIdx.x & 31;
    const int wv = threadIdx.x >> 5;
    if (lane == 0) {
#pragma unroll
        for (int i = 0; i < 5; ++i) sm[wv * 5 + i] = r[i];
    }
    __syncthreads();
    if (threadIdx.x < 5) {
        float acc = 0.0f;
#pragma unroll
        for (int w = 0; w < 8; ++w) acc += sm[w * 5 + threadIdx.x];
        part[blockIdx.x * 5 + threadIdx.x] = acc;
    }
}

__global__ __launch_bounds__(256)
void yolo_final(const float* __restrict__ part, int nblocks,
                float* __restrict__ out, float invB) {
    __shared__ float sm[5][256];
    float acc[5] = {0.f, 0.f, 0.f, 0.f, 0.f};
    for (int b = threadIdx.x; b < nblocks; b += 256) {
#pragma unroll
        for (int i = 0; i < 5; ++i) acc[i] += part[b * 5 + i];
    }
#pragma unroll
    for (int i = 0; i < 5; ++i) sm[i][threadIdx.x] = acc[i];
    __syncthreads();
    for (int off = 128; off > 0; off >>= 1) {     // fixed-order, deterministic
        if (threadIdx.x < (unsigned)off) {
#pragma unroll
            for (int i = 0; i < 5; ++i) sm[i][threadIdx.x] += sm[i][threadIdx.x + off];
        }
        __syncthreads();
    }
    if (threadIdx.x == 0) {
        float loc    = sm[0][0];
        float cobj   = sm[1][0];
        float cnoobj = sm[2][0];
        float clsum  = sm[3][0];
        float nobj   = sm[4][0];
        float cls = clsum / fmaxf(nobj, 1.0f);
        out[0] = (L_COORD * loc + cobj + L_NOOBJ * cnoobj + cls) * invB;
        out[1] = loc;
        out[2] = cobj;
        out[3] = cnoobj;
        out[4] = cls;
    }
}

extern "C" void kernel_launch(void* const* d_in, const int* in_sizes, int n_in,
                              void* d_out, int out_size, void* d_ws, size_t ws_size,
                              hipStream_t stream) {
    const float* pred = (const float*)d_in[0];
    const float* targ = (const float*)d_in[1];
    float* out  = (float*)d_out;
    float* part = (float*)d_ws;            // nblocks * 5 floats (20 KB)

    const int cells = in_sizes[0] / 30;    // B * 7 * 7
    const int nblocks = 1024;
    const float invB = 49.0f / (float)cells;   // 1 / B (exact: B = cells/49)

    yolo_partial<<<nblocks, 256, 0, stream>>>(pred, targ, part, cells);
    yolo_final<<<1, 256, 0, stream>>>(part, nblocks, out, invB);
}